// FPN_LSS_Mamba_FOBA_V2_14370960572905
// MI455X (gfx1250) — compile-verified
//
#include <hip/hip_runtime.h>
#include <cstddef>

// ---------------------------------------------------------------------------
// Types / helpers
// ---------------------------------------------------------------------------
typedef __attribute__((ext_vector_type(2))) float v2f;
typedef __attribute__((ext_vector_type(8))) float v8f;

// Global (address_space(1)) access helpers: guarantee global_load/global_store
// instead of flat_* (flat ties up DScnt + the LDS path on CDNA5).
typedef float v2f_as1 __attribute__((ext_vector_type(2), address_space(1)));
typedef float f32_as1 __attribute__((address_space(1)));

__device__ __forceinline__ v2f ldg2(const float* p) {
  return *(const v2f_as1*)(unsigned long long)p;
}
__device__ __forceinline__ void stg(float* p, float v) {
  *(f32_as1*)(unsigned long long)p = v;
}

#define LN_EPS 1e-5f

enum { ACT_NONE = 0, ACT_RELU = 1, ACT_SOFTPLUS = 2 };

__device__ __forceinline__ float act_apply(float v, int act) {
  if (act == ACT_RELU) return v > 0.f ? v : 0.f;
  if (act == ACT_SOFTPLUS) return (v > 20.f) ? v : log1pf(__expf(v));
  return v;
}

__device__ __forceinline__ int seq_to_spatial(int l, int H, int W, int col) {
  // col: seq index l = a*H + b over (W,H) grid -> spatial (h=b, w=a)
  return col ? ((l % H) * W + (l / H)) : l;
}

__device__ __forceinline__ float wave_red_sum(float v) {
  for (int o = 16; o > 0; o >>= 1) v += __shfl_down(v, o, 32);
  return v;
}

// ---------------------------------------------------------------------------
// GEMM (NT): C[M,N] = act( (A[M,K] * B[N,K]^T) * scale[n] + bias[n] )
//
// Assumptions (hold for every call in this network):
//   K % 8 == 0, lda/ldb even (aligned float2 loads), M % 16 == 0.
// Out-of-range fragment rows are CLAMPED: garbage only reaches C rows/cols
// that the store skips, so the inner loop is completely branchless.
// Wave tile 32x32 (2x2 f32 WMMA fragments), block = 4 waves -> 64x64 tile.
// ---------------------------------------------------------------------------
__global__ __launch_bounds__(128)
void gemm_nt_wmma(const float* __restrict__ A, const float* __restrict__ B,
                  float* __restrict__ C,
                  int M, int N, int K, int lda, int ldb, int ldc,
                  int revA,
                  const float* __restrict__ scale,
                  const float* __restrict__ bias,
                  int act)
{
  const int lane  = threadIdx.x & 31;
  const int wave  = threadIdx.x >> 5;
  const int m0 = blockIdx.y * 64 + (wave >> 1) * 32;
  const int n0 = blockIdx.x * 64 + (wave & 1) * 32;
  const int lr = lane & 15;
  const int kh = (lane >> 4) << 1;   // 0 or 2 (K slot inside the 16x16x4 frag)

  // Hoisted, clamped row pointers.
  const float* ap[2];
  const float* bp[2];
  #pragma unroll
  for (int i = 0; i < 2; ++i) {
    int m = m0 + i * 16 + lr;
    if (m > M - 1) m = M - 1;
    int mr = revA ? (M - 1 - m) : m;
    ap[i] = A + (size_t)mr * lda + kh;
    int n = n0 + i * 16 + lr;
    if (n > N - 1) n = N - 1;
    bp[i] = B + (size_t)n * ldb + kh;
  }

  v8f acc[2][2] = {};

  // K in chunks of 8 (2 wmma K-steps): one prefetch pair per chunk.
  for (int k0 = 0; k0 < K; k0 += 8) {
    __builtin_prefetch(ap[0] + 64, 0, 0);   // ~256B ahead
    __builtin_prefetch(bp[0] + 64, 0, 0);
    #pragma unroll
    for (int kk = 0; kk < 2; ++kk) {
      v2f af0 = ldg2(ap[0] + kk * 4);
      v2f af1 = ldg2(ap[1] + kk * 4);
      v2f bf0 = ldg2(bp[0] + kk * 4);
      v2f bf1 = ldg2(bp[1] + kk * 4);
      acc[0][0] = __builtin_amdgcn_wmma_f32_16x16x4_f32(
          false, af0, false, bf0, (short)0, acc[0][0], false, false);
      acc[0][1] = __builtin_amdgcn_wmma_f32_16x16x4_f32(
          false, af0, false, bf1, (short)0, acc[0][1], false, false);
      acc[1][0] = __builtin_amdgcn_wmma_f32_16x16x4_f32(
          false, af1, false, bf0, (short)0, acc[1][0], false, false);
      acc[1][1] = __builtin_amdgcn_wmma_f32_16x16x4_f32(
          false, af1, false, bf1, (short)0, acc[1][1], false, false);
    }
    ap[0] += 8; ap[1] += 8; bp[0] += 8; bp[1] += 8;
  }

  // Store. M % 16 == 0 -> per-fragment validity; N guarded per column.
  #pragma unroll
  for (int i = 0; i < 2; ++i) {
    const bool mval = (m0 + i * 16) < M;
    if (!mval) continue;
    const int mbase = m0 + i * 16 + ((lane >> 4) << 3);
    #pragma unroll
    for (int j = 0; j < 2; ++j) {
      int n = n0 + j * 16 + (lane & 15);
      if (n >= N) continue;
      float s = scale ? scale[n] : 1.f;
      float t = bias  ? bias[n]  : 0.f;
      #pragma unroll
      for (int rr = 0; rr < 8; ++rr)
        stg(&C[(size_t)(mbase + rr) * ldc + n], act_apply(acc[i][j][rr] * s + t, act));
    }
  }
}

// ---------------------------------------------------------------------------
// 3x3 conv as implicit GEMM over 9 taps, NHWC, on a ZERO-PADDED input
// Xp: (H+2, W+2, Cin). Wp: (Cout, 9*Cin) tap-major. Y: (H*W, Cout).
// H*W % 64 == 0, Cout % 64 == 0, Cin % 16 == 0 here -> fully branchless.
// ---------------------------------------------------------------------------
__global__ __launch_bounds__(128)
void conv3x3_wmma(const float* __restrict__ Xp, const float* __restrict__ Wp,
                  float* __restrict__ Y,
                  int H, int W, int Cin, int Cout,
                  const float* __restrict__ scale,
                  const float* __restrict__ bias,
                  int act)
{
  const int lane  = threadIdx.x & 31;
  const int wave  = threadIdx.x >> 5;
  const int m0 = blockIdx.y * 64 + (wave >> 1) * 32;
  const int n0 = blockIdx.x * 64 + (wave & 1) * 32;
  const int lr = lane & 15;
  const int kh = (lane >> 4) << 1;
  const int Wp2 = W + 2;
  const int ldw = 9 * Cin;

  const float* arow[2];
  const float* brow[2];
  #pragma unroll
  for (int i = 0; i < 2; ++i) {
    int m = m0 + i * 16 + lr;            // always < H*W (M % 64 == 0)
    int h = m / W, w = m % W;
    arow[i] = Xp + (size_t)((h + 1) * Wp2 + (w + 1)) * Cin + kh;
    int n = n0 + i * 16 + lr;            // always < Cout
    brow[i] = Wp + (size_t)n * ldw + kh;
  }

  v8f acc[2][2] = {};

  for (int tap = 0; tap < 9; ++tap) {
    const int dy = tap / 3 - 1, dx = tap % 3 - 1;
    const float* a0 = arow[0] + (ptrdiff_t)(dy * Wp2 + dx) * Cin;
    const float* a1 = arow[1] + (ptrdiff_t)(dy * Wp2 + dx) * Cin;
    const float* b0 = brow[0] + tap * Cin;
    const float* b1 = brow[1] + tap * Cin;
    for (int kc = 0; kc < Cin; kc += 16) {
      __builtin_prefetch(a0 + 64, 0, 0);
      __builtin_prefetch(b0 + 64, 0, 0);
      #pragma unroll
      for (int kk = 0; kk < 4; ++kk) {
        v2f af0 = ldg2(a0 + kk * 4);
        v2f af1 = ldg2(a1 + kk * 4);
        v2f bf0 = ldg2(b0 + kk * 4);
        v2f bf1 = ldg2(b1 + kk * 4);
        acc[0][0] = __builtin_amdgcn_wmma_f32_16x16x4_f32(
            false, af0, false, bf0, (short)0, acc[0][0], false, false);
        acc[0][1] = __builtin_amdgcn_wmma_f32_16x16x4_f32(
            false, af0, false, bf1, (short)0, acc[0][1], false, false);
        acc[1][0] = __builtin_amdgcn_wmma_f32_16x16x4_f32(
            false, af1, false, bf0, (short)0, acc[1][0], false, false);
        acc[1][1] = __builtin_amdgcn_wmma_f32_16x16x4_f32(
            false, af1, false, bf1, (short)0, acc[1][1], false, false);
      }
      a0 += 16; a1 += 16; b0 += 16; b1 += 16;
    }
  }

  #pragma unroll
  for (int i = 0; i < 2; ++i) {
    const int mbase = m0 + i * 16 + ((lane >> 4) << 3);
    #pragma unroll
    for (int j = 0; j < 2; ++j) {
      const int n = n0 + j * 16 + (lane & 15);
      const float s = scale ? scale[n] : 1.f;
      const float t = bias  ? bias[n]  : 0.f;
      #pragma unroll
      for (int rr = 0; rr < 8; ++rr)
        stg(&Y[(size_t)(mbase + rr) * Cout + n], act_apply(acc[i][j][rr] * s + t, act));
    }
  }
}

// ---------------------------------------------------------------------------
// Zero-pad NHWC (H,W,C) -> (H+2,W+2,C)
// ---------------------------------------------------------------------------
__global__ __launch_bounds__(256)
void pad_nhwc(const float* __restrict__ X, float* __restrict__ Y,
              int H, int W, int C)
{
  int idx = blockIdx.x * blockDim.x + threadIdx.x;
  int total = (H + 2) * (W + 2) * C;
  if (idx >= total) return;
  int c = idx % C, q = idx / C;
  int wp = q % (W + 2), hp = q / (W + 2);
  int h = hp - 1, w = wp - 1;
  Y[idx] = (h >= 0 && h < H && w >= 0 && w < W) ? X[(size_t)(h * W + w) * C + c] : 0.f;
}

// ---------------------------------------------------------------------------
// Row LayerNorm over last dim d. Optional second input added with reversed
// rows (combines the two scan directions: yo[l] = ys0[l] + ys1[L-1-l]).
// ---------------------------------------------------------------------------
__global__ __launch_bounds__(256)
void layernorm_rows(const float* __restrict__ X, const float* __restrict__ X2rev,
                    float* __restrict__ Y, int L, int d,
                    const float* __restrict__ gamma, const float* __restrict__ beta)
{
  __shared__ float sm1[8], sm2[8];
  const int row = blockIdx.x;
  const float* x  = X + (size_t)row * d;
  const float* x2 = X2rev ? X2rev + (size_t)(L - 1 - row) * d : nullptr;

  float s = 0.f, ss = 0.f;
  for (int i = threadIdx.x; i < d; i += 256) {
    float v = x[i] + (x2 ? x2[i] : 0.f);
    s += v; ss += v * v;
  }
  s  = wave_red_sum(s);
  ss = wave_red_sum(ss);
  const int wid = threadIdx.x >> 5, lane = threadIdx.x & 31;
  if (lane == 0) { sm1[wid] = s; sm2[wid] = ss; }
  __syncthreads();
  if (threadIdx.x == 0) {
    float a = 0.f, b = 0.f;
    for (int i = 0; i < 8; ++i) { a += sm1[i]; b += sm2[i]; }
    sm1[0] = a; sm2[0] = b;
  }
  __syncthreads();
  const float mu  = sm1[0] / d;
  const float var = sm2[0] / d - mu * mu;
  const float inv = rsqrtf(var + LN_EPS);
  for (int i = threadIdx.x; i < d; i += 256) {
    float v = x[i] + (x2 ? x2[i] : 0.f);
    Y[(size_t)row * d + i] = (v - mu) * inv * gamma[i] + beta[i];
  }
}

// ---------------------------------------------------------------------------
// Depthwise 3x3 SAME conv + bias + SiLU, NHWC.  Wd: (C,1,3,3) -> c*9 + t
// ---------------------------------------------------------------------------
__global__ __launch_bounds__(256)
void dwconv3x3_silu(const float* __restrict__ X, const float* __restrict__ Wd,
                    const float* __restrict__ Bd, float* __restrict__ Y,
                    int H, int W, int C)
{
  int idx = blockIdx.x * blockDim.x + threadIdx.x;
  int total = H * W * C;
  if (idx >= total) return;
  int c = idx % C, p = idx / C;
  int h = p / W, w = p % W;
  float acc = Bd[c];
  #pragma unroll
  for (int ky = 0; ky < 3; ++ky)
    #pragma unroll
    for (int kx = 0; kx < 3; ++kx) {
      int hy = h + ky - 1, wx = w + kx - 1;
      if (hy >= 0 && hy < H && wx >= 0 && wx < W)
        acc += X[(size_t)(hy * W + wx) * C + c] * Wd[c * 9 + ky * 3 + kx];
    }
  Y[idx] = acc / (1.f + __expf(-acc));   // SiLU
}

// ---------------------------------------------------------------------------
// seq gather / residual scatter (handles the col-major scan-order permute)
// ---------------------------------------------------------------------------
__global__ __launch_bounds__(256)
void seq_gather(const float* __restrict__ X, float* __restrict__ Y,
                int H, int W, int d, int col)
{
  int idx = blockIdx.x * blockDim.x + threadIdx.x;
  int total = H * W * d;
  if (idx >= total) return;
  int l = idx / d, c = idx % d;
  int p = seq_to_spatial(l, H, W, col);
  Y[idx] = X[(size_t)p * d + c];
}

__global__ __launch_bounds__(256)
void residual_scatter(const float* __restrict__ Yseq, float* __restrict__ Act,
                      int H, int W, int d, int col)
{
  int idx = blockIdx.x * blockDim.x + threadIdx.x;
  int total = H * W * d;
  if (idx >= total) return;
  int l = idx / d, c = idx % d;
  int p = seq_to_spatial(l, H, W, col);
  Act[(size_t)p * d + c] += Yseq[idx];
}

// ---------------------------------------------------------------------------
// Fused selective scan (one direction). 16 consecutive lanes = one channel's
// 16 states; shfl_xor reduction over n per step. Avoids materializing the
// (2,L,d,16) a/b/h tensors (~570 MB at d=640) -> pure streaming pass.
// ---------------------------------------------------------------------------
__global__ __launch_bounds__(256)
void selective_scan(const float* __restrict__ seq,   // (L,d) scan order
                    const float* __restrict__ dbl,   // (L,Wd) this direction
                    const float* __restrict__ dt,    // (L,d)  this direction
                    const float* __restrict__ A_log, // (d,16) this direction
                    const float* __restrict__ Ds,    // (d)
                    float* __restrict__ ys,          // (L,d)
                    int L, int d, int Wd, int r, int rev)
{
  const int n    = threadIdx.x & 15;
  const int cloc = threadIdx.x >> 4;            // 0..15
  const int c    = blockIdx.x * 16 + cloc;
  const bool active = (c < d);
  const float Aval = active ? -__expf(A_log[(size_t)c * 16 + n]) : 0.f;
  const float Dval = active ? Ds[c] : 0.f;
  float h = 0.f;
  for (int l = 0; l < L; ++l) {
    int lx = rev ? (L - 1 - l) : l;
    float dtv = active ? dt[(size_t)l * d + c]   : 0.f;
    float xv  = active ? seq[(size_t)lx * d + c] : 0.f;
    float Bv  = dbl[(size_t)l * Wd + r + n];
    float Cv  = dbl[(size_t)l * Wd + r + 16 + n];
    float a   = __expf(dtv * Aval);
    h = a * h + dtv * xv * Bv;
    float contrib = h * Cv;
    contrib += __shfl_xor(contrib, 1, 32);
    contrib += __shfl_xor(contrib, 2, 32);
    contrib += __shfl_xor(contrib, 4, 32);
    contrib += __shfl_xor(contrib, 8, 32);
    if (n == 0 && active)
      ys[(size_t)l * d + c] = contrib + Dval * xv;
  }
}

// ---------------------------------------------------------------------------
// Layout / misc kernels
// ---------------------------------------------------------------------------
__global__ __launch_bounds__(256)
void nchw_to_nhwc(const float* __restrict__ X, float* __restrict__ Y,
                  int C, int H, int W, int ldY, int colOff)
{
  int idx = blockIdx.x * blockDim.x + threadIdx.x;
  int total = C * H * W;
  if (idx >= total) return;
  int c = idx / (H * W), p = idx % (H * W);
  Y[(size_t)p * ldY + colOff + c] = X[idx];
}

__global__ __launch_bounds__(256)
void nhwc_to_nchw(const float* __restrict__ X, float* __restrict__ Y,
                  int C, int H, int W)
{
  int idx = blockIdx.x * blockDim.x + threadIdx.x;
  int total = C * H * W;
  if (idx >= total) return;
  int p = idx / C, c = idx % C;
  Y[(size_t)c * (H * W) + p] = X[idx];
}

__global__ __launch_bounds__(256)
void upsample_bilinear(const float* __restrict__ X, float* __restrict__ Y,
                       int Hi, int Wi, int C, int Ho, int Wo,
                       int ldY, int colOff)
{
  int idx = blockIdx.x * blockDim.x + threadIdx.x;
  int total = Ho * Wo * C;
  if (idx >= total) return;
  int c = idx % C, q = idx / C;
  int wo = q % Wo, ho = q / Wo;
  float fy = ho * ((float)(Hi - 1) / (float)(Ho - 1));
  float fx = wo * ((float)(Wi - 1) / (float)(Wo - 1));
  int y0 = (int)floorf(fy); if (y0 < 0) y0 = 0; if (y0 > Hi - 1) y0 = Hi - 1;
  int x0i = (int)floorf(fx); if (x0i < 0) x0i = 0; if (x0i > Wi - 1) x0i = Wi - 1;
  int y1 = min(y0 + 1, Hi - 1), x1i = min(x0i + 1, Wi - 1);
  float wy = fy - (float)y0, wx = fx - (float)x0i;
  float v00 = X[(size_t)(y0 * Wi + x0i) * C + c];
  float v01 = X[(size_t)(y0 * Wi + x1i) * C + c];
  float v10 = X[(size_t)(y1 * Wi + x0i) * C + c];
  float v11 = X[(size_t)(y1 * Wi + x1i) * C + c];
  float v0 = v00 * (1.f - wx) + v01 * wx;
  float v1 = v10 * (1.f - wx) + v11 * wx;
  Y[(size_t)(ho * Wo + wo) * ldY + colOff + c] = v0 * (1.f - wy) + v1 * wy;
}

__global__ __launch_bounds__(256)
void bn_prep(const float* __restrict__ g, const float* __restrict__ b,
             const float* __restrict__ m, const float* __restrict__ v,
             float* __restrict__ s, float* __restrict__ t, int C)
{
  int c = blockIdx.x * blockDim.x + threadIdx.x;
  if (c >= C) return;
  float sc = g[c] * rsqrtf(v[c] + LN_EPS);
  s[c] = sc;
  t[c] = b[c] - m[c] * sc;
}

// OIHW (Cout,Cin,3,3) -> (Cout, 9*Cin) tap-major
__global__ __launch_bounds__(256)
void repack_oihw(const float* __restrict__ Wsrc, float* __restrict__ Wdst,
                 int Cout, int Cin)
{
  int idx = blockIdx.x * blockDim.x + threadIdx.x;
  int total = Cout * 9 * Cin;
  if (idx >= total) return;
  int co = idx / (9 * Cin);
  int rem = idx % (9 * Cin);
  int tap = rem / Cin, ci = rem % Cin;
  Wdst[idx] = Wsrc[((size_t)co * Cin + ci) * 9 + tap];
}

// ---------------------------------------------------------------------------
// Host-side orchestration
// ---------------------------------------------------------------------------
struct BlkP {
  const float *ln_g, *ln_b, *in_w, *conv_w, *conv_b, *xproj_w,
              *dt_w, *dt_b, *A_log, *Ds, *on_g, *on_b, *out_w;
};

static void run_vss(const BlkP& P, float* act, int H, int W, int d, int col,
                    float* B1, float* B2, float* B3, float* B4, float* B5,
                    float* B6, float* B7, hipStream_t s)
{
  const int L  = H * W;
  const int r  = (d + 15) / 16;
  const int Wd = r + 32;
  const int tot = L * d;

  layernorm_rows<<<L, 256, 0, s>>>(act, nullptr, B1, L, d, P.ln_g, P.ln_b);

  dim3 gdd((d + 63) / 64, (L + 63) / 64);
  gemm_nt_wmma<<<gdd, 128, 0, s>>>(B1, P.in_w, B2, L, d, d, d, d, d, 0,
                                   nullptr, nullptr, ACT_NONE);

  dwconv3x3_silu<<<(tot + 255) / 256, 256, 0, s>>>(B2, P.conv_w, P.conv_b, B1, H, W, d);
  seq_gather<<<(tot + 255) / 256, 256, 0, s>>>(B1, B3, H, W, d, col);

  dim3 gx((Wd + 63) / 64, (L + 63) / 64);
  for (int k = 0; k < 2; ++k)
    gemm_nt_wmma<<<gx, 128, 0, s>>>(B3, P.xproj_w + (size_t)k * Wd * d,
                                    B4 + (size_t)k * L * Wd,
                                    L, Wd, d, d, d, Wd, k,
                                    nullptr, nullptr, ACT_NONE);

  for (int k = 0; k < 2; ++k)
    gemm_nt_wmma<<<gdd, 128, 0, s>>>(B4 + (size_t)k * L * Wd,
                                     P.dt_w + (size_t)k * d * r,
                                     B5 + (size_t)k * L * d,
                                     L, d, r, Wd, r, d, 0,
                                     nullptr, P.dt_b + (size_t)k * d, ACT_SOFTPLUS);

  for (int k = 0; k < 2; ++k)
    selective_scan<<<(d + 15) / 16, 256, 0, s>>>(B3, B4 + (size_t)k * L * Wd,
                                                 B5 + (size_t)k * L * d,
                                                 P.A_log + (size_t)k * d * 16,
                                                 P.Ds + (size_t)k * d,
                                                 k ? B7 : B6, L, d, Wd, r, k);

  layernorm_rows<<<L, 256, 0, s>>>(B6, B7, B1, L, d, P.on_g, P.on_b);

  gemm_nt_wmma<<<gdd, 128, 0, s>>>(B1, P.out_w, B2, L, d, d, d, d, d, 0,
                                   nullptr, nullptr, ACT_NONE);

  residual_scatter<<<(tot + 255) / 256, 256, 0, s>>>(B2, act, H, W, d, col);
}

static void run_conv3x3(const float* X, const float* Wsrc, float* WP, float* PAD,
                        float* Y, int H, int W, int Cin, int Cout,
                        const float* sc, const float* sh, int act, hipStream_t s)
{
  int wn = Cout * 9 * Cin;
  repack_oihw<<<(wn + 255) / 256, 256, 0, s>>>(Wsrc, WP, Cout, Cin);
  int pn = (H + 2) * (W + 2) * Cin;
  pad_nhwc<<<(pn + 255) / 256, 256, 0, s>>>(X, PAD, H, W, Cin);
  int M = H * W;
  dim3 g((Cout + 63) / 64, (M + 63) / 64);
  conv3x3_wmma<<<g, 128, 0, s>>>(PAD, WP, Y, H, W, Cin, Cout, sc, sh, act);
}

extern "C" void kernel_launch(void* const* d_in, const int* in_sizes, int n_in,
                              void* d_out, int out_size, void* d_ws, size_t ws_size,
                              hipStream_t stream)
{
  (void)in_sizes; (void)n_in; (void)out_size; (void)ws_size;
  const float* x2 = (const float*)d_in[0];   // (1,128,48,48)
  const float* x1 = (const float*)d_in[1];   // (1,512,12,12)

  BlkP blk[10];
  for (int i = 0; i < 10; ++i) {
    int b0 = 2 + i * 13;
    blk[i].ln_g    = (const float*)d_in[b0 + 0];
    blk[i].ln_b    = (const float*)d_in[b0 + 1];
    blk[i].in_w    = (const float*)d_in[b0 + 2];
    blk[i].conv_w  = (const float*)d_in[b0 + 3];
    blk[i].conv_b  = (const float*)d_in[b0 + 4];
    blk[i].xproj_w = (const float*)d_in[b0 + 5];
    blk[i].dt_w    = (const float*)d_in[b0 + 6];
    blk[i].dt_b    = (const float*)d_in[b0 + 7];
    blk[i].A_log   = (const float*)d_in[b0 + 8];
    blk[i].Ds      = (const float*)d_in[b0 + 9];
    blk[i].on_g    = (const float*)d_in[b0 + 10];
    blk[i].on_b    = (const float*)d_in[b0 + 11];
    blk[i].out_w   = (const float*)d_in[b0 + 12];
  }
  const float* c1_w = (const float*)d_in[132];
  const float* c2_w = (const float*)d_in[133];
  const float* u1_w = (const float*)d_in[134];
  const float* u2_w = (const float*)d_in[135];
  const float* u2_b = (const float*)d_in[136];
  const float* bn1g = (const float*)d_in[137];
  const float* bn1b = (const float*)d_in[138];
  const float* bn1m = (const float*)d_in[139];
  const float* bn1v = (const float*)d_in[140];
  const float* bn2g = (const float*)d_in[141];
  const float* bn2b = (const float*)d_in[142];
  const float* bn2m = (const float*)d_in[143];
  const float* bn2v = (const float*)d_in[144];
  const float* bn3g = (const float*)d_in[145];
  const float* bn3b = (const float*)d_in[146];
  const float* bn3m = (const float*)d_in[147];
  const float* bn3v = (const float*)d_in[148];

  // Workspace layout (floats)
  float* Wf = (float*)d_ws;
  size_t off = 0;
  auto alloc = [&](size_t n) { float* p = Wf + off; off += n; return p; };
  const size_t SZ_BIG = 4718592;   // 9216*512
  const size_t SZ_MED = 2359296;   // 9216*256
  float* A0  = alloc(SZ_BIG);      // main activation
  float* B1  = alloc(SZ_MED);
  float* B2  = alloc(SZ_MED);
  float* B3  = alloc(SZ_MED);
  float* B6  = alloc(SZ_MED);
  float* B7  = alloc(SZ_MED);
  float* B4  = alloc(884736);      // dbl: 2 * maxL * maxWd
  float* B5  = alloc(SZ_BIG);      // dt : 2 * maxL * maxd
  float* B8  = alloc(SZ_MED);      // second activation / conv io
  float* WP  = alloc(2949120);     // packed conv weights (max 512*9*640)
  float* PAD = alloc(4917248);     // padded conv input (max 98*98*512)
  float* BN  = alloc(2048);        // BN scale/shift

  // ---- Stage 1: x1 -> NHWC, 4x VSS @ d=512, 12x12 ----
  nchw_to_nhwc<<<(512 * 144 + 255) / 256, 256, 0, stream>>>(x1, A0, 512, 12, 12, 512, 0);
  run_vss(blk[0], A0, 12, 12, 512, 1, B1, B2, B3, B4, B5, B6, B7, stream);
  run_vss(blk[1], A0, 12, 12, 512, 0, B1, B2, B3, B4, B5, B6, B7, stream);
  run_vss(blk[2], A0, 12, 12, 512, 1, B1, B2, B3, B4, B5, B6, B7, stream);
  run_vss(blk[3], A0, 12, 12, 512, 0, B1, B2, B3, B4, B5, B6, B7, stream);

  // ---- Upsample x4 + concat with x2 -> (48,48,640) NHWC in B8 ----
  upsample_bilinear<<<(48 * 48 * 512 + 255) / 256, 256, 0, stream>>>(
      A0, B8, 12, 12, 512, 48, 48, 640, 128);
  nchw_to_nhwc<<<(128 * 2304 + 255) / 256, 256, 0, stream>>>(x2, B8, 128, 48, 48, 640, 0);

  // ---- Stage 2: 4x VSS @ d=640, 48x48 ----
  run_vss(blk[4], B8, 48, 48, 640, 1, B1, B2, B3, B4, B5, B6, B7, stream);
  run_vss(blk[5], B8, 48, 48, 640, 0, B1, B2, B3, B4, B5, B6, B7, stream);
  run_vss(blk[6], B8, 48, 48, 640, 1, B1, B2, B3, B4, B5, B6, B7, stream);
  run_vss(blk[7], B8, 48, 48, 640, 0, B1, B2, B3, B4, B5, B6, B7, stream);

  // ---- Conv head: c1 + bn1 + relu, c2 + bn2 + relu ----
  bn_prep<<<2, 256, 0, stream>>>(bn1g, bn1b, bn1m, bn1v, BN, BN + 512, 512);
  run_conv3x3(B8, c1_w, WP, PAD, A0, 48, 48, 640, 512, BN, BN + 512, ACT_RELU, stream);
  bn_prep<<<2, 256, 0, stream>>>(bn2g, bn2b, bn2m, bn2v, BN, BN + 512, 512);
  run_conv3x3(A0, c2_w, WP, PAD, B1, 48, 48, 512, 512, BN, BN + 512, ACT_RELU, stream);

  // ---- Upsample x2 -> (96,96,512), u1 + bn3 + relu, u2 (1x1 gemm + bias) ----
  upsample_bilinear<<<(96 * 96 * 512 + 255) / 256, 256, 0, stream>>>(
      B1, A0, 48, 48, 512, 96, 96, 512, 0);
  bn_prep<<<1, 256, 0, stream>>>(bn3g, bn3b, bn3m, bn3v, BN, BN + 256, 256);
  run_conv3x3(A0, u1_w, WP, PAD, B8, 96, 96, 512, 256, BN, BN + 256, ACT_RELU, stream);
  dim3 gu((256 + 63) / 64, (9216 + 63) / 64);
  gemm_nt_wmma<<<gu, 128, 0, stream>>>(B8, u2_w, A0, 9216, 256, 256, 256, 256, 256, 0,
                                       nullptr, u2_b, ACT_NONE);

  // ---- Stage 3: 2x VSS @ d=256, 96x96 ----
  run_vss(blk[8], A0, 96, 96, 256, 1, B1, B2, B3, B4, B5, B6, B7, stream);
  run_vss(blk[9], A0, 96, 96, 256, 0, B1, B2, B3, B4, B5, B6, B7, stream);

  // ---- Output: NHWC -> NCHW ----
  nhwc_to_nchw<<<(9216 * 256 + 255) / 256, 256, 0, stream>>>(
      A0, (float*)d_out, 256, 96, 96);
}